// QubitEmbedding_17617955848415
// MI455X (gfx1250) — compile-verified
//
#include <hip/hip_runtime.h>
#include <math.h>

typedef __attribute__((ext_vector_type(2))) float v2f;
typedef __attribute__((ext_vector_type(4))) float v4f;
typedef __attribute__((ext_vector_type(8))) float v8f;

#define XWPS 288   // pair-row stride (floats) for pair-interleaved xw layout
#define CS   68    // LDS row stride (floats) for coeff tile (272B = 17*16B)

// LDS byte offset of a pointer that actually points into shared memory
static __device__ __forceinline__ unsigned lds_off(const void* p) {
    return (unsigned)(size_t)(__attribute__((address_space(3))) const char*)p;
}

// ---------------------------------------------------------------- zero scratch
__global__ void k_zero(float* __restrict__ p, long long n4) {
    long long i = (long long)blockIdx.x * blockDim.x + threadIdx.x;
    if (i < n4) reinterpret_cast<v4f*>(p)[i] = (v4f){0.f, 0.f, 0.f, 0.f};
}

// ---------------------------------------------------------------- in-degree (exact, integer atomics)
__global__ void k_deg(const int* __restrict__ dst, unsigned* __restrict__ deg, int E) {
    int e = blockIdx.x * blockDim.x + threadIdx.x;
    if (e < E) atomicAdd(&deg[dst[e]], 1u);
}

// ---------------------------------------------------------------- dinv = deg>0 ? rsqrt(deg) : 0   (in place)
__global__ void k_dinv(void* __restrict__ degbuf, int N) {
    int i = blockIdx.x * blockDim.x + threadIdx.x;
    if (i < N) {
        int c = reinterpret_cast<int*>(degbuf)[i];
        reinterpret_cast<float*>(degbuf)[i] = (c > 0) ? rsqrtf((float)c) : 0.0f;
    }
}

// ---------------------------------------------------------------- coeff[dst][src%Q] += dinv[src]  (1 atomic/edge)
__global__ void k_coeff(const int* __restrict__ src, const int* __restrict__ dst,
                        const float* __restrict__ dinv, float* __restrict__ coeff,
                        int E, int Q) {
    int e = blockIdx.x * blockDim.x + threadIdx.x;
    if (e < E) {
        unsigned s = (unsigned)src[e];
        unsigned d = (unsigned)dst[e];
        float w = dinv[s];
        unsafeAtomicAdd(&coeff[(size_t)d * (unsigned)Q + (s % (unsigned)Q)], w);
    }
}

// ---------------------------------------------------------------- xw_small = emb[0:Q] @ W, written pair-interleaved:
//   element (k,c) -> xwp[(k>>1)*XWPS + 2*c + (k&1)]   (assumes H==128, Q==64)
__global__ void k_xw(const float* __restrict__ emb, const float* __restrict__ W,
                     float* __restrict__ xwp, int Q, int H) {
    int idx = blockIdx.x * blockDim.x + threadIdx.x;
    if (idx < Q * H) {
        int r = idx / H, c = idx % H;
        float acc = 0.0f;
        for (int k = 0; k < H; ++k) acc = fmaf(emb[r * H + k], W[k * H + c], acc);
        xwp[(r >> 1) * XWPS + 2 * c + (r & 1)] = acc;
    }
}

// ---------------------------------------------------------------- pe per node-group g (= node/Q):  pe[g%S][h]
__global__ void k_pe(float* __restrict__ pe, const int* __restrict__ dS, int G, int H) {
    int idx = blockIdx.x * blockDim.x + threadIdx.x;
    if (idx < G * H) {
        int g = idx / H, h = idx % H;
        int S = dS[0];
        int s = (S > 0) ? (g % S) : 0;
        float div = expf((float)(h & ~1) * (-logf(10000.0f) / (float)H));
        float ang = (float)s * div;
        pe[idx] = (h & 1) ? cosf(ang) : sinf(ang);
    }
}

// ---------------------------------------------------------------- main: out = dinv[n]*(coeff[n,:]@xw) + bias + pe
// Block = 128 threads (4 waves), owns 64 consecutive nodes; H=128, K=Q=64 hard-wired.
__global__ __launch_bounds__(128) void k_main(const float* __restrict__ coeff,
                                              const float* __restrict__ xwp,
                                              const float* __restrict__ dinv,
                                              const float* __restrict__ bias,
                                              const float* __restrict__ pegrp,
                                              float* __restrict__ out) {
    __shared__ __align__(16) float s_xwp[32 * XWPS];  // 36864 B, pair-interleaved
    __shared__ __align__(16) float s_cf[64 * CS];     // 17408 B, padded rows
    __shared__ float s_dinv[64];
    __shared__ float s_add[128];

    const int tid = threadIdx.x;
    const size_t blk = blockIdx.x;
    const size_t nodeBase = blk * 64;

    // ---- async stage xw pair-layout (linear 36864B copy, 16B chunks) ----
    {
        const unsigned lbase = lds_off(s_xwp);
        const unsigned long long gbase = (unsigned long long)xwp;
        for (int i = tid; i < (32 * XWPS) / 4; i += 128) {
            unsigned l = lbase + (unsigned)i * 16u;
            unsigned long long g = gbase + (unsigned long long)i * 16ull;
            asm volatile("global_load_async_to_lds_b128 %0, %1, off"
                         :: "v"(l), "v"(g) : "memory");
        }
    }
    // ---- async stage this block's 64x64 coeff tile into padded rows ----
    {
        const unsigned lbase = lds_off(s_cf);
        const unsigned long long gbase = (unsigned long long)(coeff + nodeBase * 64);
        for (int i = tid; i < 1024; i += 128) {          // 1024 chunks of 16B
            int row = i >> 4, w = i & 15;
            unsigned l = lbase + (unsigned)(row * (CS * 4) + w * 16);
            unsigned long long g = gbase + (unsigned long long)(row * 256 + w * 16);
            asm volatile("global_load_async_to_lds_b128 %0, %1, off"
                         :: "v"(l), "v"(g) : "memory");
        }
    }
    if (tid < 64)  s_dinv[tid] = dinv[nodeBase + tid];
    if (tid < 128) s_add[tid]  = bias[tid] + pegrp[blk * 128 + tid];
    asm volatile("s_wait_asynccnt 0" ::: "memory");
    __syncthreads();

    const int lane = tid & 31;
    const int wave = tid >> 5;
    const int rowBase = wave * 16;
    const int m16 = lane & 15;     // row (A) / col (B,C,D) within tile
    const int kh  = lane >> 4;     // which K-pair half

    // A fragments: 16 K-steps of 16x4 f32 (coeff rows), one b64 LDS load each
    v2f a[16];
#pragma unroll
    for (int t = 0; t < 16; ++t) {
        int off = (rowBase + m16) * CS + 4 * t + 2 * kh;   // even -> 8B aligned
        a[t] = *reinterpret_cast<const v2f*>(&s_cf[off]);
    }

#pragma unroll
    for (int nt = 0; nt < 8; ++nt) {
        const int col = nt * 16 + m16;
        v8f acc = {0.f, 0.f, 0.f, 0.f, 0.f, 0.f, 0.f, 0.f};
#pragma unroll
        for (int t = 0; t < 16; ++t) {
            const int kp = 2 * t + kh;                       // pair-row = kk>>1
            // B fragment {xw[kk][col], xw[kk+1][col]} is contiguous: one b64 load
            v2f b = *reinterpret_cast<const v2f*>(&s_xwp[kp * XWPS + 2 * col]);
            acc = __builtin_amdgcn_wmma_f32_16x16x4_f32(
                false, a[t], false, b, (short)0, acc, false, false);
        }
        const int m0 = kh * 8;
#pragma unroll
        for (int j = 0; j < 8; ++j) {
            const int m = rowBase + m0 + j;
            float v = s_dinv[m] * acc[j] + s_add[col];
            out[(nodeBase + (size_t)m) * 128 + col] = v;
        }
    }
}

// ---------------------------------------------------------------- host launcher
extern "C" void kernel_launch(void* const* d_in, const int* in_sizes, int n_in,
                              void* d_out, int out_size, void* d_ws, size_t ws_size,
                              hipStream_t stream) {
    const float* emb  = (const float*)d_in[0];
    const float* W    = (const float*)d_in[1];
    const float* bias = (const float*)d_in[2];
    const int*   esrc = (const int*)d_in[3];
    const int*   edst = (const int*)d_in[4];
    const int*   dS   = (const int*)d_in[6];   // num_slices (device scalar)

    const int    H = in_sizes[2];              // 128
    const int    Q = in_sizes[0] / H - 1;      // 64
    const int    E = in_sizes[3];              // 2*N
    const size_t N = (size_t)out_size / (size_t)H;   // 524288
    const size_t G = N / (size_t)Q;            // batch*slices = 8192

    // workspace layout (all 256B-aligned by construction)
    char*  ws     = (char*)d_ws;
    float* coeff  = (float*)ws;                                   // N*Q floats (134 MB)
    size_t coeffN = N * (size_t)Q;
    void*  degbuf = (void*)(ws + coeffN * 4);                     // N floats (deg -> dinv in place)
    float* xwp    = (float*)(ws + (coeffN + N) * 4);              // 32*XWPS floats, pair-interleaved
    float* pe     = (float*)(ws + (coeffN + N) * 4 + (size_t)(32 * XWPS) * 4); // G*H floats
    float* out    = (float*)d_out;

    // 1) zero coeff + deg  (N*(Q+1) floats; divisible by 4)
    long long n4 = (long long)((coeffN + N) / 4);
    k_zero<<<(unsigned)((n4 + 255) / 256), 256, 0, stream>>>((float*)ws, n4);

    // 2) in-degree
    k_deg<<<(E + 255) / 256, 256, 0, stream>>>(edst, (unsigned*)degbuf, E);

    // 3) dinv
    k_dinv<<<(unsigned)((N + 255) / 256), 256, 0, stream>>>(degbuf, (int)N);

    // 4) coeff scatter (1 float atomic per edge)
    k_coeff<<<(E + 255) / 256, 256, 0, stream>>>(esrc, edst, (const float*)degbuf, coeff, E, Q);

    // 5) xw_small = emb[0:Q] @ W  (pair-interleaved output layout)
    k_xw<<<(Q * H + 255) / 256, 256, 0, stream>>>(emb, W, xwp, Q, H);

    // 6) positional encoding table per node-group
    k_pe<<<(unsigned)((G * (size_t)H + 255) / 256), 256, 0, stream>>>(pe, dS, (int)G, H);

    // 7) main WMMA GEMM + fused epilogue (64 nodes per block)
    k_main<<<(unsigned)(N / 64), 128, 0, stream>>>(coeff, xwp, (const float*)degbuf, bias, pe, out);
}